// RNAOptimizationModel_33663953666492
// MI455X (gfx1250) — compile-verified
//
#include <hip/hip_runtime.h>
#include <hip/hip_bf16.h>

typedef __attribute__((ext_vector_type(2))) float v2f;
typedef __attribute__((ext_vector_type(8))) float v8f;

#define N_UNITS 8
#define M_ATOMS 1024
#define NPTS (N_UNITS * M_ATOMS)
#define NPAIRS 28                        // C(8,2) unit pairs with ui < uj
#define BLOCKS_PER_PAIR 16
#define PAIR_GRID (NPAIRS * BLOCKS_PER_PAIR)   // 448 blocks, 8 waves each
#define TILES_PER_WAVE 32                // each wave: half a tile-row (64x64 tiles per pair)

#define LOG2E 1.44269504088896340736f

// accum[0] = sum of |p|^2 over all 8192 points (for L1)
// accum[1] = sum of exp(1 - d2) over masked pairs (for L_int)

__global__ __launch_bounds__(32) void init_kernel(float* accum) {
    if (threadIdx.x < 2) accum[threadIdx.x] = 0.0f;
}

// rowPts[p] = (2L*x, 2L*y, 2L*z, -L*sq)    -> WMMA A operand (K=0..3)
// colPts[p] = (x,    y,    z,    L*(1-sq)) -> WMMA B operand (K3 = 1.0 const) + C broadcast
__global__ __launch_bounds__(256) void prep_kernel(const float* __restrict__ positions,
                                                   const float* __restrict__ euler,
                                                   const float* __restrict__ coords,
                                                   float4* __restrict__ rowPts,
                                                   float4* __restrict__ colPts,
                                                   float* __restrict__ accum) {
    int p = blockIdx.x * 256 + threadIdx.x;     // 0 .. 8191
    int u = p >> 10;                            // unit
    int a = p & (M_ATOMS - 1);                  // atom

    float phi   = euler[u * 3 + 0];
    float theta = euler[u * 3 + 1];
    float psi   = euler[u * 3 + 2];
    float cp = cosf(phi),   sp = sinf(phi);
    float ct = cosf(theta), st = sinf(theta);
    float cs = cosf(psi),   ss = sinf(psi);

    // R = Rz(psi) * Ry(theta) * Rx(phi)
    float r00 = cs * ct;
    float r01 = cs * st * sp - ss * cp;
    float r02 = cs * st * cp + ss * sp;
    float r10 = ss * ct;
    float r11 = ss * st * sp + cs * cp;
    float r12 = ss * st * cp - cs * sp;
    float r20 = -st;
    float r21 = ct * sp;
    float r22 = ct * cp;

    float cx = coords[a * 3 + 0];
    float cy = coords[a * 3 + 1];
    float cz = coords[a * 3 + 2];

    float tx = r00 * cx + r01 * cy + r02 * cz + positions[u * 3 + 0];
    float ty = r10 * cx + r11 * cy + r12 * cz + positions[u * 3 + 1];
    float tz = r20 * cx + r21 * cy + r22 * cz + positions[u * 3 + 2];
    float sq = tx * tx + ty * ty + tz * tz;

    rowPts[p] = make_float4(2.0f * LOG2E * tx, 2.0f * LOG2E * ty,
                            2.0f * LOG2E * tz, -LOG2E * sq);
    colPts[p] = make_float4(tx, ty, tz, LOG2E * (1.0f - sq));

    // wave32 reduction of sq for L1, one atomic per wave
    float s = sq;
#pragma unroll
    for (int off = 16; off > 0; off >>= 1) s += __shfl_down(s, off, 32);
    if ((threadIdx.x & 31) == 0) atomicAdd(&accum[0], s);
}

__global__ __launch_bounds__(256) void pair_kernel(const float4* __restrict__ rowPts,
                                                   const float4* __restrict__ colPts,
                                                   float* __restrict__ accum) {
    const int lane = threadIdx.x & 31;
    const int hi   = lane >> 4;   // 0: K=0,1 half, 1: K=2,3 half
    const int m    = lane & 15;   // row/col within 16-wide tile

    // Pair index is SGPR-uniform (function of blockIdx only) -> pure scalar mapping loop
    int pairIdx     = blockIdx.x >> 4;          // 0..27
    int blockInPair = blockIdx.x & 15;          // 0..15
    int k = pairIdx, ui = 0, rem = N_UNITS - 1;
    while (k >= rem) { k -= rem; ++ui; --rem; }
    int uj = ui + 1 + k;

    // 128 waves per pair: wave owns tile-row ti, half of the 64 column tiles
    int waveInPair = blockInPair * 8 + (threadIdx.x >> 5);  // 0..127
    int ti     = waveInPair >> 1;                           // 0..63
    int tjbase = (waveInPair & 1) << 5;                     // 0 or 32

    const float4* rowTile = rowPts + (ui << 10) + (ti << 4);
    const float4* colTile = colPts + (uj << 10) + (tjbase << 4);

    // A operand is loop-invariant: load + pack once
    float4 ra = rowTile[m];
    v2f A;
    A.x = hi ? ra.z : ra.x;               // 2L*z     | 2L*x
    A.y = hi ? ra.w : ra.y;               // -L*sq_i  | 2L*y

    float sum0 = 0.0f, sum1 = 0.0f;

    for (int it = 0; it < TILES_PER_WAVE; ++it) {
        float4 cb = colTile[it * 16 + m];  // next 16 column points

        v2f B;
        B.x = hi ? cb.z : cb.x;            // z        | x
        B.y = hi ? 1.0f : cb.y;            // 1.0      | y

        // C broadcast: every output row of column N gets L*(1 - sq_j)
        float cval = cb.w;
        v8f c = {cval, cval, cval, cval, cval, cval, cval, cval};

        // D = A*B + C = L * (2*dot - sq_i + 1 - sq_j) = log2(e) * (1 - d2)
        c = __builtin_amdgcn_wmma_f32_16x16x4_f32(
                /*neg_a=*/false, A, /*neg_b=*/false, B,
                /*c_mod=*/(short)0, c, /*reuse_a=*/false, /*reuse_b=*/false);

#pragma unroll
        for (int e = 0; e < 8; e += 2) {
            sum0 += __builtin_amdgcn_exp2f(c[e]);      // exp(1 - d2) via raw v_exp_f32
            sum1 += __builtin_amdgcn_exp2f(c[e + 1]);
        }
    }

    float sum = sum0 + sum1;
#pragma unroll
    for (int off = 16; off > 0; off >>= 1) sum += __shfl_down(sum, off, 32);
    if (lane == 0) atomicAdd(&accum[1], sum);
}

__global__ __launch_bounds__(32) void finalize_kernel(const float* __restrict__ positions,
                                                      const float* __restrict__ accum,
                                                      float* __restrict__ out) {
    if (threadIdx.x == 0) {
        float cx = 0.0f, cy = 0.0f, cz = 0.0f;
#pragma unroll
        for (int u = 0; u < N_UNITS; ++u) {
            cx += positions[u * 3 + 0];
            cy += positions[u * 3 + 1];
            cz += positions[u * 3 + 2];
        }
        float L1   = accum[0] * (1.0f / (float)N_UNITS);
        float Lcom = cx * cx + cy * cy + cz * cz;     // ALPHA = 1
        float Lint = 0.5f * accum[1];                 // LAMBDA1 = 0.5
        out[0] = L1 + Lcom + Lint;
    }
}

extern "C" void kernel_launch(void* const* d_in, const int* in_sizes, int n_in,
                              void* d_out, int out_size, void* d_ws, size_t ws_size,
                              hipStream_t stream) {
    const float* positions = (const float*)d_in[0];   // (8,3)
    const float* euler     = (const float*)d_in[1];   // (8,3)
    const float* coords    = (const float*)d_in[2];   // (1024,3)
    float* out = (float*)d_out;

    float*  accum  = (float*)d_ws;
    float4* rowPts = (float4*)((char*)d_ws + 256);                         // 128 KB
    float4* colPts = (float4*)((char*)d_ws + 256 + NPTS * sizeof(float4)); // 128 KB

    init_kernel<<<1, 32, 0, stream>>>(accum);
    prep_kernel<<<NPTS / 256, 256, 0, stream>>>(positions, euler, coords,
                                                rowPts, colPts, accum);
    pair_kernel<<<PAIR_GRID, 256, 0, stream>>>(rowPts, colPts, accum);
    finalize_kernel<<<1, 32, 0, stream>>>(positions, accum, out);
}